// PeriodicBoundary_19129784336915
// MI455X (gfx1250) — compile-verified
//
#include <hip/hip_runtime.h>

// CDNA5 / gfx1250 periodic neighbour list, three-pass (count / prefix / emit).
// Pairwise-distance engine: V_WMMA_F32_16X16X4_F32
//   d2 = |a|^2 - 2 a.q + |q|^2  ==  A(16x4) x B(4x16) + C(16x16)
//   A row = ( xa, ya, za, 1 );  B col = ( -2xq, -2yq, -2zq, |q|^2 );  C = |a|^2
// One wave owns a 16-row x 16-candidate tile; A and C are loop-invariant.
// Pass 1 counts with per-lane accumulators + shfl butterfly (no ballots);
// pass 3 uses ballot/prefix packing because emission order matters.

typedef float v2f __attribute__((ext_vector_type(2)));
typedef float v8f __attribute__((ext_vector_type(8)));

#define CUTOFF2   0.09f
#define NCELLS    27
#define CELLS_PAD 32          // padded stride for cnts/offs arrays
#define SELF_CELL 13
#define MASKVAL   (-1)

__global__ void pb_zero_max(int* out_max) { *out_max = 0; }

// ---------------- shared device helpers ----------------

__device__ __forceinline__ void pb_stage(const float* __restrict__ pos,
                                         float* xs, float* ys, float* zs, int N) {
    for (int i = threadIdx.x; i < N; i += blockDim.x) {
        xs[i] = pos[3 * i + 0];
        ys[i] = pos[3 * i + 1];
        zs[i] = pos[3 * i + 2];
    }
}

__device__ __forceinline__ void pb_make_AC(const float* xs, const float* ys,
                                           const float* zs, int rowbase, int l16,
                                           int hi, int N, v2f& A, v8f& C) {
    int arow = rowbase + l16; if (arow >= N) arow = N - 1;
    A.x = hi ? zs[arow] : xs[arow];
    A.y = hi ? 1.0f     : ys[arow];
#pragma unroll
    for (int v = 0; v < 8; ++v) {
        int r = rowbase + v + 8 * hi; if (r >= N) r = N - 1;
        const float px = xs[r], py = ys[r], pz = zs[r];
        C[v] = fmaf(px, px, fmaf(py, py, pz * pz));
    }
}

__device__ __forceinline__ void pb_shift(const float* __restrict__ cellm, int c,
                                         int& sx, int& sy, int& sz,
                                         float& shx, float& shy, float& shz) {
    // x-fastest ordering: matches meshgrid(ij).T.reshape(-1,3)
    sx = (c % 3) - 1; sy = ((c / 3) % 3) - 1; sz = (c / 9) - 1;
    const float fx = (float)sx, fy = (float)sy, fz = (float)sz;
    shx = fx * cellm[0] + fy * cellm[3] + fz * cellm[6];
    shy = fx * cellm[1] + fy * cellm[4] + fz * cellm[7];
    shz = fx * cellm[2] + fy * cellm[5] + fz * cellm[8];
}

__device__ __forceinline__ v8f pb_tile_d2(const float* xs, const float* ys,
                                          const float* zs, int col, int hi,
                                          float shx, float shy, float shz,
                                          v2f A, v8f Cinit) {
    const float qx = xs[col] + shx;
    const float qy = ys[col] + shy;
    const float qz = zs[col] + shz;
    const float qsq = fmaf(qx, qx, fmaf(qy, qy, qz * qz));
    v2f B;                                        // branchless half-select
    B.x = hi ? (-2.0f * qz) : (-2.0f * qx);
    B.y = hi ? qsq          : (-2.0f * qy);
    return __builtin_amdgcn_wmma_f32_16x16x4_f32(
        false, A, false, B, (short)0, Cinit, false, false);
}

// ---------------- pass 1: count hits per (row, cell) ----------------
// grid = (N/16 rowtiles, ceil(27/8) cell groups), 256 thr = 8 waves; wave -> one cell.
// No ordering needed -> per-lane accumulators, one butterfly reduce at the end.

__global__ __launch_bounds__(256)
void pb_count(const float* __restrict__ pos, const float* __restrict__ cellm,
              int* __restrict__ cnts, int N) {
    extern __shared__ float smem[];
    float* xs = smem; float* ys = smem + N; float* zs = smem + 2 * N;

    const int tid = threadIdx.x, lane = tid & 31, wave = tid >> 5;
    const int hi = (lane >= 16) ? 1 : 0, l16 = lane & 15;

    pb_stage(pos, xs, ys, zs, N);
    __syncthreads();

    const int cell = blockIdx.y * 8 + wave;
    if (cell >= NCELLS) return;
    const int rowbase = blockIdx.x * 16;

    v2f A; v8f Cinit;
    pb_make_AC(xs, ys, zs, rowbase, l16, hi, N, A, Cinit);
    int sx, sy, sz; float shx, shy, shz;
    pb_shift(cellm, cell, sx, sy, sz, shx, shy, shz);
    const bool selfcell = (cell == SELF_CELL);

    int cnt[8];
#pragma unroll
    for (int v = 0; v < 8; ++v) cnt[v] = 0;

    const int ntiles = N >> 4;
#pragma unroll 2
    for (int t = 0; t < ntiles; ++t) {
        const int col = (t << 4) + l16;
        v8f D = pb_tile_d2(xs, ys, zs, col, hi, shx, shy, shz, A, Cinit);
#pragma unroll
        for (int v = 0; v < 8; ++v) {
            const int row = rowbase + v + 8 * hi;
            const bool pred = (D[v] <= CUTOFF2) && (!selfcell || col != row)
                              && (row < N);
            cnt[v] += pred ? 1 : 0;               // per-lane, no cross-lane op
        }
    }

    // butterfly sum within each 16-lane half (xor 1,2,4,8 stays inside a half)
#pragma unroll
    for (int d = 1; d < 16; d <<= 1) {
#pragma unroll
        for (int v = 0; v < 8; ++v) cnt[v] += __shfl_xor(cnt[v], d);
    }

    if (l16 == 0) {                               // lane0 -> rows v, lane16 -> rows v+8
        const int rb = rowbase + 8 * hi;
#pragma unroll
        for (int v = 0; v < 8; ++v)
            if (rb + v < N) cnts[(size_t)(rb + v) * CELLS_PAD + cell] = cnt[v];
    }
}

// ---------------- pass 2: per-row exclusive prefix over cells, tail fill, max ----

__global__ __launch_bounds__(256)
void pb_prefix(const int* __restrict__ cnts, int* __restrict__ offs,
               int* __restrict__ neigh, int* __restrict__ cellidx,
               int* __restrict__ out_max, int N, int K) {
    const int tid = threadIdx.x, lane = tid & 31, wave = tid >> 5;
    const int row = blockIdx.x * 8 + wave;
    if (row >= N) return;

    const int v = (lane < NCELLS) ? cnts[(size_t)row * CELLS_PAD + lane] : 0;
    int x = v;
#pragma unroll
    for (int d = 1; d < 32; d <<= 1) {
        const int y = __shfl_up(x, d);
        if (lane >= d) x += y;
    }
    if (lane < NCELLS) offs[(size_t)row * CELLS_PAD + lane] = x - v; // exclusive
    const int total = __shfl(x, 31);
    if (lane == 0) atomicMax(out_max, total);

    const int start = (total < K) ? total : K;
    for (int p = start + lane; p < K; p += 32) {
        neigh[(size_t)row * K + p] = MASKVAL;
        int* ci = cellidx + ((size_t)row * K + p) * 3;
        ci[0] = MASKVAL; ci[1] = MASKVAL; ci[2] = MASKVAL;
    }
}

// ---------------- pass 3: emit hits at precomputed offsets ----------------

__global__ __launch_bounds__(256)
void pb_emit(const float* __restrict__ pos, const float* __restrict__ cellm,
             const int* __restrict__ offs, int* __restrict__ neigh,
             int* __restrict__ cellidx, int N, int K) {
    extern __shared__ float smem[];
    float* xs = smem; float* ys = smem + N; float* zs = smem + 2 * N;
    int* cnt = (int*)(smem + 3 * N);              // 16 counters per wave

    const int tid = threadIdx.x, lane = tid & 31, wave = tid >> 5;
    const int hi = (lane >= 16) ? 1 : 0, l16 = lane & 15;

    pb_stage(pos, xs, ys, zs, N);

    const int cell = blockIdx.y * 8 + wave;
    const int rowbase = blockIdx.x * 16;
    const bool active = (cell < NCELLS);
    int myoff = K;
    if (active && lane < 16) {
        const int r = rowbase + lane;
        myoff = (r < N) ? offs[(size_t)r * CELLS_PAD + cell] : K;
        cnt[wave * 16 + lane] = myoff;
    }
    __syncthreads();
    if (!active) return;
    if (__ballot(lane < 16 && myoff < K) == 0ull) return;  // whole task past cap

    int* mycnt = cnt + wave * 16;

    v2f A; v8f Cinit;
    pb_make_AC(xs, ys, zs, rowbase, l16, hi, N, A, Cinit);
    int sx, sy, sz; float shx, shy, shz;
    pb_shift(cellm, cell, sx, sy, sz, shx, shy, shz);
    const bool selfcell = (cell == SELF_CELL);

    const int ntiles = N >> 4;
#pragma unroll 2
    for (int t = 0; t < ntiles; ++t) {
        const int col = (t << 4) + l16;
        v8f D = pb_tile_d2(xs, ys, zs, col, hi, shx, shy, shz, A, Cinit);
#pragma unroll
        for (int v = 0; v < 8; ++v) {
            const int row = rowbase + v + 8 * hi;
            const bool pred = (D[v] <= CUTOFF2) && (!selfcell || col != row)
                              && (row < N);
            const unsigned mask = (unsigned)__ballot(pred);
            const unsigned half = hi ? (mask >> 16) : (mask & 0xFFFFu);
            const int base = mycnt[v + 8 * hi];
            if (pred) {
                const int p = base + (int)__popc(half & ((1u << l16) - 1u));
                if (p < K) {
                    neigh[(size_t)row * K + p] = col;
                    int* ci = cellidx + ((size_t)row * K + p) * 3;
                    ci[0] = sx; ci[1] = sy; ci[2] = sz;
                }
            }
            if (l16 == 0)
                mycnt[v + 8 * hi] = base + (int)__popc(half);
        }
    }
}

// ---------------- fallback: fused single-pass (small ws_size) ----------------

__global__ __launch_bounds__(256)
void pb_fused(const float* __restrict__ pos, const float* __restrict__ cellm,
              int* __restrict__ neigh, int* __restrict__ cellidx,
              int* __restrict__ out_max, int N, int K) {
    extern __shared__ float smem[];
    float* xs = smem; float* ys = smem + N; float* zs = smem + 2 * N;
    int* cnt = (int*)(smem + 3 * N);

    const int tid = threadIdx.x, lane = tid & 31, wave = tid >> 5;
    const int nwaves = blockDim.x >> 5;
    const int hi = (lane >= 16) ? 1 : 0, l16 = lane & 15;

    pb_stage(pos, xs, ys, zs, N);
    if (tid < nwaves * 16) cnt[tid] = 0;
    __syncthreads();

    const int rowbase = blockIdx.x * (nwaves * 16) + wave * 16;
    if (rowbase >= N) return;
    int* mycnt = cnt + wave * 16;

    v2f A; v8f Cinit;
    pb_make_AC(xs, ys, zs, rowbase, l16, hi, N, A, Cinit);
    const int ntiles = N >> 4;

    for (int c = 0; c < NCELLS; ++c) {
        int sx, sy, sz; float shx, shy, shz;
        pb_shift(cellm, c, sx, sy, sz, shx, shy, shz);
        const bool selfcell = (c == SELF_CELL);
#pragma unroll 2
        for (int t = 0; t < ntiles; ++t) {
            const int col = (t << 4) + l16;
            v8f D = pb_tile_d2(xs, ys, zs, col, hi, shx, shy, shz, A, Cinit);
#pragma unroll
            for (int v = 0; v < 8; ++v) {
                const int row = rowbase + v + 8 * hi;
                const bool pred = (D[v] <= CUTOFF2) && (!selfcell || col != row)
                                  && (row < N);
                const unsigned mask = (unsigned)__ballot(pred);
                const unsigned half = hi ? (mask >> 16) : (mask & 0xFFFFu);
                const int base = mycnt[v + 8 * hi];
                if (pred) {
                    const int p = base + (int)__popc(half & ((1u << l16) - 1u));
                    if (p < K) {
                        neigh[(size_t)row * K + p] = col;
                        int* ci = cellidx + ((size_t)row * K + p) * 3;
                        ci[0] = sx; ci[1] = sy; ci[2] = sz;
                    }
                }
                if (l16 == 0)
                    mycnt[v + 8 * hi] = base + (int)__popc(half);
            }
        }
    }
#pragma unroll 1
    for (int v = 0; v < 16; ++v) {
        const int row = rowbase + v;
        if (row >= N) break;
        const int total = mycnt[v];
        const int start = (total < K) ? total : K;
        for (int p = start + lane; p < K; p += 32) {
            neigh[(size_t)row * K + p] = MASKVAL;
            int* ci = cellidx + ((size_t)row * K + p) * 3;
            ci[0] = MASKVAL; ci[1] = MASKVAL; ci[2] = MASKVAL;
        }
        if (lane == 0) atomicMax(out_max, total);
    }
}

// ---------------- host launch ----------------

extern "C" void kernel_launch(void* const* d_in, const int* in_sizes, int n_in,
                              void* d_out, int out_size, void* d_ws, size_t ws_size,
                              hipStream_t stream) {
    const float* pos   = (const float*)d_in[0];   // (N,3) f32
    const float* cellm = (const float*)d_in[1];   // (3,3) f32
    // d_in[2] = max_neighbours (K derived from out_size)

    const int N = in_sizes[0] / 3;
    const int K = (out_size - 1) / (4 * N);       // out = N*K + 3*N*K + 1 ints

    int* neigh   = (int*)d_out;
    int* cellidx = neigh + (size_t)N * K;
    int* out_max = cellidx + (size_t)N * K * 3;

    pb_zero_max<<<1, 1, 0, stream>>>(out_max);

    const size_t needed = (size_t)N * CELLS_PAD * sizeof(int) * 2;
    const size_t pos_lds = (size_t)(3 * N) * sizeof(float);

    if (ws_size >= needed && (N % 16) == 0) {
        int* cnts = (int*)d_ws;
        int* offs = cnts + (size_t)N * CELLS_PAD;
        dim3 gridp(N / 16, (NCELLS + 7) / 8);      // 3456 wave-tasks
        pb_count <<<gridp, 256, pos_lds, stream>>>(pos, cellm, cnts, N);
        pb_prefix<<<(N + 7) / 8, 256, 0, stream>>>(cnts, offs, neigh, cellidx,
                                                   out_max, N, K);
        pb_emit  <<<gridp, 256, pos_lds + 8 * 16 * sizeof(int), stream>>>(
            pos, cellm, offs, neigh, cellidx, N, K);
    } else {
        const int threads = 256;                   // 8 waves -> 128 rows/block
        const int rpb = (threads / 32) * 16;
        const int blocks = (N + rpb - 1) / rpb;
        pb_fused<<<blocks, threads, pos_lds + (threads / 32) * 16 * sizeof(int),
                   stream>>>(pos, cellm, neigh, cellidx, out_max, N, K);
    }
}